// OmniFluids2D_74406013435961
// MI455X (gfx1250) — compile-verified
//
#include <hip/hip_runtime.h>
#include <math.h>

#define BB    4
#define NXD   256
#define NYD   256
#define NFEAT 5
#define KEXP  4
#define WID   64
#define LLAY  4

typedef __attribute__((ext_vector_type(2))) float v2f;
typedef __attribute__((ext_vector_type(8))) float v8f;

__device__ __forceinline__ float gelu_f(float v) {
    return 0.5f * v * (1.0f + erff(v * 0.70710678118654752f));
}

// ---------------------------------------------------------------------------
// Generic strided batched GEMM using V_WMMA_F32_16X16X4_F32 (full f32).
// C[b1,b2][m,n] (+)= alpha * sum_k A[b1,b2][m,k] * B[b1,b2][k,n]
// One wave computes a 16x64 strip (4 accumulators); the A fragment is loaded
// once per k-step and reused by 4 WMMAs.  N must be a multiple of 64.
// ---------------------------------------------------------------------------
struct GemmDesc {
    const float* A; const float* B; float* C;
    int M, N, K;
    long lar, lac, lbr, lbc, lcr, lcc;
    int nb2;
    long sa1, sa2, sb1, sb2, sc1, sc2;
    float alpha; int acc;
};

__global__ void __launch_bounds__(32) k_gemm(GemmDesc g) {
    int bz = blockIdx.z;
    int b2 = bz % g.nb2, b1 = bz / g.nb2;
    const float* A = g.A + (long)b1 * g.sa1 + (long)b2 * g.sa2;
    const float* B = g.B + (long)b1 * g.sb1 + (long)b2 * g.sb2;
    float*       C = g.C + (long)b1 * g.sc1 + (long)b2 * g.sc2;
    int lane = threadIdx.x;
    int m0 = blockIdx.y * 16, n0 = blockIdx.x * 64;
    int mr = lane & 15;
    int ks = (lane < 16) ? 0 : 2;          // K-pair select per lane half
    v8f acc0 = {}, acc1 = {}, acc2 = {}, acc3 = {};
    const float* Arow = A + (long)(m0 + mr) * g.lar;
    long bcol = (long)(n0 + mr) * g.lbc;
    long nst  = 16L * g.lbc;               // column-tile stride in B

    if (g.lac == 1) {
        // unit-stride A (channel mixing): single aligned 8-byte A load
        for (int kk = 0; kk < g.K; kk += 4) {
            v2f a = *(const v2f*)(Arow + kk + ks);
            long br0 = (long)(kk + ks) * g.lbr, br1 = br0 + g.lbr;
            v2f b;
            b.x = B[br0 + bcol];           b.y = B[br1 + bcol];
            acc0 = __builtin_amdgcn_wmma_f32_16x16x4_f32(false, a, false, b, (short)0, acc0, false, false);
            b.x = B[br0 + bcol + nst];     b.y = B[br1 + bcol + nst];
            acc1 = __builtin_amdgcn_wmma_f32_16x16x4_f32(false, a, false, b, (short)0, acc1, false, false);
            b.x = B[br0 + bcol + 2*nst];   b.y = B[br1 + bcol + 2*nst];
            acc2 = __builtin_amdgcn_wmma_f32_16x16x4_f32(false, a, false, b, (short)0, acc2, false, false);
            b.x = B[br0 + bcol + 3*nst];   b.y = B[br1 + bcol + 3*nst];
            acc3 = __builtin_amdgcn_wmma_f32_16x16x4_f32(false, a, false, b, (short)0, acc3, false, false);
        }
    } else {
        for (int kk = 0; kk < g.K; kk += 4) {
            v2f a;
            a.x = Arow[(long)(kk + ks)     * g.lac];
            a.y = Arow[(long)(kk + ks + 1) * g.lac];
            long br0 = (long)(kk + ks) * g.lbr, br1 = br0 + g.lbr;
            v2f b;
            b.x = B[br0 + bcol];           b.y = B[br1 + bcol];
            acc0 = __builtin_amdgcn_wmma_f32_16x16x4_f32(false, a, false, b, (short)0, acc0, false, false);
            b.x = B[br0 + bcol + nst];     b.y = B[br1 + bcol + nst];
            acc1 = __builtin_amdgcn_wmma_f32_16x16x4_f32(false, a, false, b, (short)0, acc1, false, false);
            b.x = B[br0 + bcol + 2*nst];   b.y = B[br1 + bcol + 2*nst];
            acc2 = __builtin_amdgcn_wmma_f32_16x16x4_f32(false, a, false, b, (short)0, acc2, false, false);
            b.x = B[br0 + bcol + 3*nst];   b.y = B[br1 + bcol + 3*nst];
            acc3 = __builtin_amdgcn_wmma_f32_16x16x4_f32(false, a, false, b, (short)0, acc3, false, false);
        }
    }
    int rowadd = (lane < 16) ? 0 : 8;
    v8f accs[4] = {acc0, acc1, acc2, acc3};
#pragma unroll
    for (int nt = 0; nt < 4; nt++) {
#pragma unroll
        for (int r = 0; r < 8; r++) {
            long ci = (long)(m0 + r + rowadd) * g.lcr
                    + (long)(n0 + nt * 16 + mr) * g.lcc;
            float v = g.alpha * accs[nt][r];
            C[ci] = g.acc ? (C[ci] + v) : v;
        }
    }
}

static inline void gemm(hipStream_t s, const float* A, const float* B, float* C,
                        int M, int N, int K,
                        long lar, long lac, long lbr, long lbc, long lcr, long lcc,
                        int nb1, int nb2,
                        long sa1, long sa2, long sb1, long sb2, long sc1, long sc2,
                        float alpha, int acc) {
    GemmDesc g;
    g.A = A; g.B = B; g.C = C; g.M = M; g.N = N; g.K = K;
    g.lar = lar; g.lac = lac; g.lbr = lbr; g.lbc = lbc; g.lcr = lcr; g.lcc = lcc;
    g.nb2 = nb2; g.sa1 = sa1; g.sa2 = sa2; g.sb1 = sb1; g.sb2 = sb2;
    g.sc1 = sc1; g.sc2 = sc2; g.alpha = alpha; g.acc = acc;
    dim3 grid(N / 64, M / 16, nb1 * nb2);
    k_gemm<<<grid, dim3(32), 0, s>>>(g);
}

// ---------------------------------------------------------------------------
// Trig transform matrices (DFT / DST expressed as dense transforms, ortho norm,
// truncated to 64 modes).  Fy:[y][m][2]  Iy:[part][m][y]  Fx:[x][k][2]
// Ix:[part][k][x]
// ---------------------------------------------------------------------------
__global__ void k_init_trig(float* Fy, float* Iy, float* Fx, float* Ix) {
    int idx = blockIdx.x * blockDim.x + threadIdx.x;
    if (idx >= 4 * 32768) return;
    int which = idx >> 15;
    int r = idx & 32767;
    const double PI2 = 6.283185307179586476925286766559;
    if (which == 0) {               // forward y: rfft/sqrt(256)
        int y = r >> 7; int rest = r & 127; int m = rest >> 1; int part = rest & 1;
        double a = PI2 * (double)((m * y) % NYD) / (double)NYD;
        Fy[r] = part ? (float)(-sin(a) / 16.0) : (float)(cos(a) / 16.0);
    } else if (which == 1) {        // inverse y (64 modes -> 256)
        int part = r >> 14; int q = r & 16383; int m = q >> 8; int y = q & 255;
        double a = PI2 * (double)((m * y) % NYD) / (double)NYD;
        float v;
        if (part == 0) v = (m == 0) ? (1.0f / 16.0f) : (float)(2.0 * cos(a) / 16.0);
        else           v = (m == 0) ? 0.0f : (float)(-2.0 * sin(a) / 16.0);
        Iy[r] = v;
    } else if (which == 2) {        // forward DST (odd-ext length-510 rfft)
        int x = r >> 7; int rest = r & 127; int k = rest >> 1; int part = rest & 1;
        double s = 1.0 / sqrt(510.0);
        float v = 0.0f;
        if (part == 0) {
            if (x == 0)        v = (float)s;
            else if (x == 255) v = (k & 1) ? (float)(-s) : (float)s;
        } else {
            if (x >= 1 && x <= 254) {
                double a = PI2 * (double)((k * x) % 510) / 510.0;
                v = (float)(-2.0 * sin(a) * s);
            }
        }
        Fx[r] = v;
    } else {                        // inverse DST (64 modes -> first 256 of irfft510)
        int part = r >> 14; int q = r & 16383; int k = q >> 8; int x = q & 255;
        double s = 1.0 / sqrt(510.0);
        double a = PI2 * (double)((k * x) % 510) / 510.0;
        float v;
        if (part == 0) v = (k == 0) ? (float)s : (float)(2.0 * cos(a) * s);
        else           v = (k == 0) ? 0.0f : (float)(-2.0 * sin(a) * s);
        Ix[r] = v;
    }
}

// Input lift: h = gelu([x, gx, gy] @ in_w + in_b)
__global__ void k_lift(const float* __restrict__ x, const float* __restrict__ in_w,
                       const float* __restrict__ in_b, float* __restrict__ h) {
    long idx = (long)blockIdx.x * blockDim.x + threadIdx.x; // B*NX*NY*64
    int c = (int)(idx & 63);
    long row = idx >> 6;
    int y = (int)(row & 255);
    int xc = (int)((row >> 8) & 255);
    float gx = 6.283185307179586f * (float)xc / 256.0f;
    float gy = 6.283185307179586f * (float)y / 256.0f;
    float s = in_b[c];
    const float* xr = x + row * NFEAT;
#pragma unroll
    for (int f = 0; f < NFEAT; f++) s += xr[f] * in_w[f * WID + c];
    s += gx * in_w[5 * WID + c] + gy * in_w[6 * WID + c];
    h[row * WID + c] = gelu_f(s);
}

// Routing MLP + softmax(./10): att[b][k]
__global__ void k_fnu(const float* __restrict__ params,
                      const float* __restrict__ w1, const float* __restrict__ b1,
                      const float* __restrict__ w2, const float* __restrict__ b2,
                      const float* __restrict__ w3, const float* __restrict__ b3,
                      float* __restrict__ att) {
    __shared__ float a1[BB][128], a2[BB][128], lg[BB][KEXP];
    int t = threadIdx.x;
    for (int b = 0; b < BB; b++) {
        float s = b1[t];
        for (int p = 0; p < 8; p++) s += params[b * 8 + p] * w1[p * 128 + t];
        a1[b][t] = gelu_f(s);
    }
    __syncthreads();
    for (int b = 0; b < BB; b++) {
        float s = b2[t];
        for (int q = 0; q < 128; q++) s += a1[b][q] * w2[q * 128 + t];
        a2[b][t] = gelu_f(s);
    }
    __syncthreads();
    if (t < BB * KEXP) {
        int b = t / KEXP, k = t % KEXP;
        float s = b3[k];
        for (int q = 0; q < 128; q++) s += a2[b][q] * w3[q * KEXP + k];
        lg[b][k] = s * 0.1f;   // / TSOFT
    }
    __syncthreads();
    if (t < BB) {
        float mx = lg[t][0];
        for (int k = 1; k < KEXP; k++) mx = fmaxf(mx, lg[t][k]);
        float den = 0.f, e[KEXP];
        for (int k = 0; k < KEXP; k++) { e[k] = expf(lg[t][k] - mx); den += e[k]; }
        for (int k = 0; k < KEXP; k++) att[t * KEXP + k] = e[k] / den;
    }
}

// w[b,i,o,m,2] = sum_k att[b,k] * fw[k,i,o,m,2]
__global__ void k_mixw(const float* __restrict__ att, const float* __restrict__ fw,
                       float* __restrict__ wout) {
    long idx = (long)blockIdx.x * blockDim.x + threadIdx.x; // 2,097,152
    const long per_b = 64L * 64 * 64 * 2;
    int b = (int)(idx / per_b);
    long rem = idx % per_b;
    float s = 0.f;
#pragma unroll
    for (int k = 0; k < KEXP; k++) s += att[b * KEXP + k] * fw[(long)k * per_b + rem];
    wout[idx] = s;
}

// Fused FeedForward (64->256 relu ->64) + LayerNorm; rb read then overwritten
// with backcast b; h += b.  4 waves, 16 rows per block, both GEMMs on WMMA f32.
__global__ void __launch_bounds__(128) k_ff(float* rb, float* __restrict__ h,
        const float* __restrict__ w0, const float* __restrict__ b0,
        const float* __restrict__ w1, const float* __restrict__ b1v,
        const float* __restrict__ g, const float* __restrict__ be) {
    __shared__ float sR[16][64];
    __shared__ float sH[16][256];
    __shared__ float sT[16][64];
    __shared__ float sMu[16], sRs[16];
    int t = threadIdx.x, lane = t & 31, wv = t >> 5;
    long rowbase = (long)blockIdx.x * 16;
    for (int i = t; i < 16 * 64; i += 128) sR[i >> 6][i & 63] = rb[rowbase * 64 + i];
    __syncthreads();
    int mr = lane & 15, ks = (lane < 16) ? 0 : 2, rowadd = (lane < 16) ? 0 : 8;
    // GEMM1: hidden = relu(sR @ w0 + b0); wave wv covers cols [64wv,64wv+64)
    {
        int nb = wv * 64;
        v8f a0 = {}, a1 = {}, a2 = {}, a3 = {};
        for (int kk = 0; kk < 64; kk += 4) {
            v2f a, b;
            a.x = sR[mr][kk + ks]; a.y = sR[mr][kk + ks + 1];
            const float* B0 = w0 + (kk + ks) * 256 + nb + mr;
            const float* B1 = B0 + 256;
            b.x = B0[0];  b.y = B1[0];
            a0 = __builtin_amdgcn_wmma_f32_16x16x4_f32(false, a, false, b, (short)0, a0, false, false);
            b.x = B0[16]; b.y = B1[16];
            a1 = __builtin_amdgcn_wmma_f32_16x16x4_f32(false, a, false, b, (short)0, a1, false, false);
            b.x = B0[32]; b.y = B1[32];
            a2 = __builtin_amdgcn_wmma_f32_16x16x4_f32(false, a, false, b, (short)0, a2, false, false);
            b.x = B0[48]; b.y = B1[48];
            a3 = __builtin_amdgcn_wmma_f32_16x16x4_f32(false, a, false, b, (short)0, a3, false, false);
        }
        v8f accs[4] = {a0, a1, a2, a3};
#pragma unroll
        for (int nt = 0; nt < 4; nt++) {
#pragma unroll
            for (int rr = 0; rr < 8; rr++) {
                int col = nb + nt * 16 + mr;
                sH[rr + rowadd][col] = fmaxf(accs[nt][rr] + b0[col], 0.0f);
            }
        }
    }
    __syncthreads();
    // GEMM2: out = sH @ w1 + b1; wave wv covers cols [16wv,16wv+16)
    {
        int n0 = wv * 16;
        v8f acc = {};
        for (int kk = 0; kk < 256; kk += 4) {
            v2f a, b;
            a.x = sH[mr][kk + ks]; a.y = sH[mr][kk + ks + 1];
            b.x = w1[(kk + ks) * 64 + n0 + mr];
            b.y = w1[(kk + ks + 1) * 64 + n0 + mr];
            acc = __builtin_amdgcn_wmma_f32_16x16x4_f32(false, a, false, b,
                                                        (short)0, acc, false, false);
        }
#pragma unroll
        for (int rr = 0; rr < 8; rr++) sT[rr + rowadd][n0 + mr] = acc[rr] + b1v[n0 + mr];
    }
    __syncthreads();
    if (t < 16) {
        float mu = 0.f;
        for (int c = 0; c < 64; c++) mu += sT[t][c];
        mu *= (1.0f / 64.0f);
        float var = 0.f;
        for (int c = 0; c < 64; c++) { float d = sT[t][c] - mu; var += d * d; }
        var *= (1.0f / 64.0f);
        sMu[t] = mu; sRs[t] = rsqrtf(var + 1e-5f);
    }
    __syncthreads();
    for (int i = t; i < 16 * 64; i += 128) {
        int rr = i >> 6, c = i & 63;
        float val = (sT[rr][c] - sMu[rr]) * sRs[rr] * g[c] + be[c];
        long gi = (rowbase + rr) * 64 + c;
        rb[gi] = val;          // backcast b
        h[gi] += val;
    }
}

// Pack [fa_w | fb_w | 0pad] -> Wc[f][64][80], biases -> bc[f][80]
__global__ void k_init_headw(const float* __restrict__ fa_w, const float* __restrict__ fa_b,
                             const float* __restrict__ fb_w, const float* __restrict__ fb_b,
                             float* __restrict__ Wc, float* __restrict__ bc) {
    int idx = blockIdx.x * blockDim.x + threadIdx.x;
    if (idx < NFEAT * 64 * 80) {
        int f = idx / (64 * 80); int rem = idx % (64 * 80);
        int c = rem / 80; int j = rem % 80;
        float v = 0.f;
        if (j < 36)      v = fa_w[(f * 64 + c) * 36 + j];
        else if (j < 70) v = fb_w[(f * 64 + c) * 34 + (j - 36)];
        Wc[idx] = v;
    }
    if (idx < NFEAT * 80) {
        int f = idx / 80, j = idx % 80;
        float v = 0.f;
        if (j < 36)      v = fa_b[f * 36 + j];
        else if (j < 70) v = fb_b[f * 34 + j - 36];
        bc[idx] = v;
    }
}

// Heads: ha = gelu(gelu(b) @ Wc + bc); conv1(8ch,k12,s2)+gelu; conv2(k12,s2);
// out = x0 + u*dt.  5 waves, 16 rows per block, GEMM on WMMA f32.
__global__ void __launch_bounds__(160) k_heads(const float* __restrict__ bbuf,
        const float* __restrict__ x0, const float* __restrict__ Wc,
        const float* __restrict__ bc, const float* __restrict__ c1w,
        const float* __restrict__ c1b, const float* __restrict__ c2w,
        const float* __restrict__ c2b, float* __restrict__ out) {
    __shared__ float sA[16][64];
    __shared__ float sHA[16][80];
    __shared__ float sU[16][8][30];
    int t = threadIdx.x, lane = t & 31, wv = t >> 5;
    int f = blockIdx.y;
    long rowbase = (long)blockIdx.x * 16;
    for (int i = t; i < 16 * 64; i += 160)
        sA[i >> 6][i & 63] = gelu_f(bbuf[rowbase * 64 + i]);
    __syncthreads();
    int mr = lane & 15, ks = (lane < 16) ? 0 : 2, rowadd = (lane < 16) ? 0 : 8;
    {
        int n0 = wv * 16;
        const float* Wf = Wc + (long)f * 64 * 80;
        v8f acc = {};
        for (int kk = 0; kk < 64; kk += 4) {
            v2f a, b;
            a.x = sA[mr][kk + ks]; a.y = sA[mr][kk + ks + 1];
            b.x = Wf[(kk + ks) * 80 + n0 + mr];
            b.y = Wf[(kk + ks + 1) * 80 + n0 + mr];
            acc = __builtin_amdgcn_wmma_f32_16x16x4_f32(false, a, false, b,
                                                        (short)0, acc, false, false);
        }
#pragma unroll
        for (int rr = 0; rr < 8; rr++) {
            int col = n0 + mr;
            sHA[rr + rowadd][col] = gelu_f(acc[rr] + bc[f * 80 + col]);
        }
    }
    __syncthreads();
    for (int v = t; v < 16 * 8 * 30; v += 160) {
        int r = v / 240, rem = v % 240, c8 = rem / 30, p = rem % 30;
        float s = c1b[f * 8 + c8];
        const float* wk = c1w + (f * 8 + c8) * 12;
#pragma unroll
        for (int kk = 0; kk < 12; kk++) s += sHA[r][2 * p + kk] * wk[kk];
        sU[r][c8][p] = gelu_f(s);
    }
    __syncthreads();
    {
        int r = t / 10, p = t % 10;
        float s = c2b[f];
#pragma unroll
        for (int c8 = 0; c8 < 8; c8++) {
            const float* wk = c2w + (f * 8 + c8) * 12;
#pragma unroll
            for (int kk = 0; kk < 12; kk++) s += sU[r][c8][2 * p + kk] * wk[kk];
        }
        long row = rowbase + r;
        float dt = (float)(p + 1) * 0.1f;
        out[(row * NFEAT + f) * 10 + p] = x0[row * NFEAT + f] + s * dt;
    }
}

// ---------------------------------------------------------------------------
extern "C" void kernel_launch(void* const* d_in, const int* in_sizes, int n_in,
                              void* d_out, int out_size, void* d_ws, size_t ws_size,
                              hipStream_t stream) {
    (void)in_sizes; (void)n_in; (void)out_size; (void)ws_size;
    const float* x      = (const float*)d_in[0];
    const float* params = (const float*)d_in[1];
    const float* in_w   = (const float*)d_in[2];
    const float* in_b   = (const float*)d_in[3];
    const float* fnu_w1 = (const float*)d_in[4];
    const float* fnu_b1 = (const float*)d_in[5];
    const float* fnu_w2 = (const float*)d_in[6];
    const float* fnu_b2 = (const float*)d_in[7];
    const float* fnu_w3 = (const float*)d_in[8];
    const float* fnu_b3 = (const float*)d_in[9];
    const float* fw_y   = (const float*)d_in[10];
    const float* fw_x   = (const float*)d_in[11];
    const float* ff_w0  = (const float*)d_in[12];
    const float* ff_b0  = (const float*)d_in[13];
    const float* ff_w1  = (const float*)d_in[14];
    const float* ff_b1  = (const float*)d_in[15];
    const float* ln_g   = (const float*)d_in[16];
    const float* ln_b   = (const float*)d_in[17];
    const float* fa_w   = (const float*)d_in[18];
    const float* fa_b   = (const float*)d_in[19];
    const float* fb_w   = (const float*)d_in[20];
    const float* fb_b   = (const float*)d_in[21];
    const float* c1_w   = (const float*)d_in[22];
    const float* c1_b   = (const float*)d_in[23];
    const float* c2_w   = (const float*)d_in[24];
    const float* c2_b   = (const float*)d_in[25];
    float* outp = (float*)d_out;

    float* ws  = (float*)d_ws;
    float* h   = ws;                      // [b][x][y][c]      16,777,216
    float* bb  = h  + 16777216;           // R / backcast b    16,777,216
    float* z   = bb + 16777216;           // zy/zx [b][p][m][x|y][c]  8,388,608
    float* ly  = z  + 8388608;            // lowy [b][p][m][x][o]     8,388,608
    float* lx  = ly + 8388608;            // lowx [b][p][kx][y][o]    8,388,608
    float* wy  = lx + 8388608;            // [b][i][o][m][2]          2,097,152
    float* wx  = wy + 2097152;
    float* Fy  = wx + 2097152;            // 32768 each
    float* Iy  = Fy + 32768;
    float* Fx  = Iy + 32768;
    float* Ix  = Fx + 32768;
    float* att = Ix + 32768;              // 16
    float* Wc  = att + 16;                // 25600
    float* bc  = Wc + 25600;              // 400

    const long HB = (long)NXD * NYD * 64;     // h batch stride (b): 4,194,304
    const long ZB = 2L * 64 * 256 * 64;       // z/ly/lx batch stride (b): 2,097,152
    const long ZP = 64L * 256 * 64;           // part offset: 1,048,576
    const long WB = 64L * 64 * 64 * 2;        // wy/wx batch stride (b): 524,288

    k_init_trig<<<512, 256, 0, stream>>>(Fy, Iy, Fx, Ix);
    k_init_headw<<<100, 256, 0, stream>>>(fa_w, fa_b, fb_w, fb_b, Wc, bc);
    k_lift<<<65536, 256, 0, stream>>>(x, in_w, in_b, h);

    for (int l = 0; l < LLAY; l++) {
        k_fnu<<<1, 128, 0, stream>>>(params, fnu_w1 + l * 1024, fnu_b1 + l * 128,
                                     fnu_w2 + l * 16384, fnu_b2 + l * 128,
                                     fnu_w3 + l * 512, fnu_b3 + l * 4, att);
        k_mixw<<<8192, 256, 0, stream>>>(att, fw_y + (long)l * 2097152, wy);
        k_mixw<<<8192, 256, 0, stream>>>(att, fw_x + (long)l * 2097152, wx);

        // ---- forward Y transform: zy[b,p,m,x,c] = sum_y Fy[y,m,p]*h[b,x,y,c]
        for (int p = 0; p < 2; p++)
            gemm(stream, Fy + p, h, z + p * ZP, 64, 64, 256,
                 2, 128, 64, 1, 16384, 1,
                 4, 256, 0, 0, HB, (long)NYD * 64, ZB, 64, 1.f, 0);
        // ---- mixing Y (complex): ly = zy * wy, per (b,m), rows x, K=i
        gemm(stream, z,      wy,     ly,      256, 64, 64, 64, 1, 8192, 128, 64, 1,
             4, 64, ZB, 16384, WB, 2, ZB, 16384,  1.f, 0);
        gemm(stream, z + ZP, wy + 1, ly,      256, 64, 64, 64, 1, 8192, 128, 64, 1,
             4, 64, ZB, 16384, WB, 2, ZB, 16384, -1.f, 1);
        gemm(stream, z,      wy + 1, ly + ZP, 256, 64, 64, 64, 1, 8192, 128, 64, 1,
             4, 64, ZB, 16384, WB, 2, ZB, 16384,  1.f, 0);
        gemm(stream, z + ZP, wy,     ly + ZP, 256, 64, 64, 64, 1, 8192, 128, 64, 1,
             4, 64, ZB, 16384, WB, 2, ZB, 16384,  1.f, 1);
        // ---- inverse Y into R (=bb): R[b,x,y,o] = sum_m Iy[p,m,y]*ly[b,p,m,x,o]
        gemm(stream, Iy,         ly,      bb, 256, 64, 64, 1, 256, 16384, 1, 64, 1,
             4, 256, 0, 0, ZB, 64, HB, (long)NYD * 64, 1.f, 0);
        gemm(stream, Iy + 16384, ly + ZP, bb, 256, 64, 64, 1, 256, 16384, 1, 64, 1,
             4, 256, 0, 0, ZB, 64, HB, (long)NYD * 64, 1.f, 1);

        // ---- forward X (DST): zx[b,p,kx,y,c] = sum_x Fx[x,kx,p]*h[b,x,y,c]
        for (int p = 0; p < 2; p++)
            gemm(stream, Fx + p, h, z + p * ZP, 64, 64, 256,
                 2, 128, (long)NYD * 64, 1, 16384, 1,
                 4, 256, 0, 0, HB, 64, ZB, 64, 1.f, 0);
        // ---- mixing X (complex): lx = zx * wx, per (b,kx), rows y, K=i
        gemm(stream, z,      wx,     lx,      256, 64, 64, 64, 1, 8192, 128, 64, 1,
             4, 64, ZB, 16384, WB, 2, ZB, 16384,  1.f, 0);
        gemm(stream, z + ZP, wx + 1, lx,      256, 64, 64, 64, 1, 8192, 128, 64, 1,
             4, 64, ZB, 16384, WB, 2, ZB, 16384, -1.f, 1);
        gemm(stream, z,      wx + 1, lx + ZP, 256, 64, 64, 64, 1, 8192, 128, 64, 1,
             4, 64, ZB, 16384, WB, 2, ZB, 16384,  1.f, 0);
        gemm(stream, z + ZP, wx,     lx + ZP, 256, 64, 64, 64, 1, 8192, 128, 64, 1,
             4, 64, ZB, 16384, WB, 2, ZB, 16384,  1.f, 1);
        // ---- inverse X accumulated into R: R[b,x,y,o] += sum_kx Ix[p,kx,x]*lx[...]
        gemm(stream, Ix,         lx,      bb, 256, 64, 64, 1, 256, 16384, 1,
             (long)NYD * 64, 1, 4, 256, 0, 0, ZB, 64, HB, 64, 1.f, 1);
        gemm(stream, Ix + 16384, lx + ZP, bb, 256, 64, 64, 1, 256, 16384, 1,
             (long)NYD * 64, 1, 4, 256, 0, 0, ZB, 64, HB, 64, 1.f, 1);

        // ---- FeedForward + LayerNorm (reads R from bb, writes b into bb, h += b)
        k_ff<<<16384, 128, 0, stream>>>(bb, h,
                                        ff_w0 + l * 16384, ff_b0 + l * 256,
                                        ff_w1 + l * 16384, ff_b1 + l * 64,
                                        ln_g + l * 64, ln_b + l * 64);
    }

    // heads on gelu(last backcast)
    k_heads<<<dim3(16384, 5), 160, 0, stream>>>(bb, x, Wc, bc,
                                                c1_w, c1_b, c2_w, c2_b, outp);
}